// ProteinFeatures_3504693314242
// MI455X (gfx1250) — compile-verified
//
#include <hip/hip_runtime.h>
#include <hip/hip_bf16.h>
#include <float.h>
#include <math.h>

typedef __attribute__((ext_vector_type(2))) float v2f;
typedef __attribute__((ext_vector_type(8))) float v8f;

#define BB   4
#define LL   2048
#define KNN_ 32
#define EF_  128
#define XPITCH 132   // padded row pitch (floats): 132 % 64 == 4 -> conflict-free b64 reads

// ---------------------------------------------------------------------------
// Kernel 1: top-K (smallest-32, ascending) per row of masked dist_ca.
// One 256-thread block per (b,i) row. Row cached in LDS, 32 argmin rounds.
// ---------------------------------------------------------------------------
__global__ __launch_bounds__(256) void pf_topk_kernel(
    const float* __restrict__ dist, const float* __restrict__ mask_angle,
    float* __restrict__ eidx_out_f, int* __restrict__ idx_ws,
    float* __restrict__ dn_ws)
{
    __shared__ float vals[LL];
    __shared__ float rv[8];
    __shared__ int   ri[8];
    __shared__ float sbcast;

    const int row  = blockIdx.x;
    const int tid  = threadIdx.x;
    const int wid  = tid >> 5;
    const int lane = tid & 31;

    const float* __restrict__ drow = dist       + (size_t)row * LL;
    const float* __restrict__ mrow = mask_angle + (size_t)row * LL;

    float dm[8];
    float lmax = -FLT_MAX;
    #pragma unroll
    for (int t = 0; t < 8; ++t) {
        int j = tid + 256 * t;
        float mk = mrow[j];
        float v  = mk * drow[j];
        vals[j]  = v;
        dm[t]    = mk;
        lmax     = fmaxf(lmax, v);
    }
    #pragma unroll
    for (int o = 16; o >= 1; o >>= 1) lmax = fmaxf(lmax, __shfl_xor(lmax, o, 32));
    if (lane == 0) rv[wid] = lmax;
    __syncthreads();
    if (tid == 0) {
        float mx = rv[0];
        #pragma unroll
        for (int w = 1; w < 8; ++w) mx = fmaxf(mx, rv[w]);
        sbcast = mx;
    }
    __syncthreads();
    const float Dmax = sbcast;
    #pragma unroll
    for (int t = 0; t < 8; ++t) {
        int j = tid + 256 * t;
        vals[j] += (1.0f - dm[t]) * Dmax;
    }
    __syncthreads();

    for (int r = 0; r < KNN_; ++r) {
        float bv = FLT_MAX;
        int   bi = 0x7fffffff;
        #pragma unroll
        for (int t = 0; t < 8; ++t) {
            int j = tid + 256 * t;
            float v = vals[j];
            if (v < bv || (v == bv && j < bi)) { bv = v; bi = j; }
        }
        #pragma unroll
        for (int o = 16; o >= 1; o >>= 1) {
            float ov = __shfl_xor(bv, o, 32);
            int   oi = __shfl_xor(bi, o, 32);
            if (ov < bv || (ov == bv && oi < bi)) { bv = ov; bi = oi; }
        }
        if (lane == 0) { rv[wid] = bv; ri[wid] = bi; }
        __syncthreads();
        if (tid == 0) {
            float v = rv[0]; int ix = ri[0];
            #pragma unroll
            for (int w = 1; w < 8; ++w)
                if (rv[w] < v || (rv[w] == v && ri[w] < ix)) { v = rv[w]; ix = ri[w]; }
            size_t o = (size_t)row * KNN_ + r;
            eidx_out_f[o] = (float)ix;
            idx_ws[o]     = ix;
            dn_ws[o]      = v;
            vals[ix] = FLT_MAX;
        }
        __syncthreads();
    }
}

// ---------------------------------------------------------------------------
// Kernel 2: edge features -> 128x128 f32 matvec via V_WMMA_F32_16X16X4_F32
// -> LayerNorm -> E.  4 waves/block, 1 wave = 16 edges (half a node's kNN).
// edge_W transposed into LDS (Wt[n][k], pitch 132); per-wave 16x132 A-tile.
// ---------------------------------------------------------------------------
__global__ __launch_bounds__(128) void pf_edge_kernel(
    const float* __restrict__ omega, const float* __restrict__ theta,
    const float* __restrict__ phi,
    const int* __restrict__ residue_idx, const int* __restrict__ chain_enc,
    const float* __restrict__ pos_W, const float* __restrict__ pos_b,
    const float* __restrict__ edge_W,
    const float* __restrict__ ln_g, const float* __restrict__ ln_b,
    const int* __restrict__ idx_ws, const float* __restrict__ dn_ws,
    float* __restrict__ outE)
{
    __shared__ float Wt[EF_ * XPITCH];        // 67584 B
    __shared__ float xs[4 * 16 * XPITCH];     // 33792 B

    const int tid = threadIdx.x;

    // Transpose-load edge_W[k][n] (row-major [128,128]) -> Wt[n*XPITCH + k]
    #pragma unroll 4
    for (int k = 0; k < EF_; ++k)
        Wt[tid * XPITCH + k] = edge_W[k * EF_ + tid];
    __syncthreads();

    const int w    = tid >> 5;
    const int lane = tid & 31;
    const int m    = lane & 15;
    const int half = lane >> 4;

    const int group  = blockIdx.x * 4 + w;     // 16-edge group
    const int e_base = group * 16;
    const int node   = e_base / KNN_;          // same node for whole group
    const int b      = node / LL;

    const size_t e = (size_t)e_base + m;       // this lane's edge (both halves)
    const int j    = idx_ws[e];
    float* __restrict__ xrow = &xs[(w * 16 + m) * XPITCH];

    const size_t pair = (size_t)node * LL + j;
    const int ri_i = residue_idx[node];
    const int ri_j = residue_idx[(size_t)b * LL + j];
    const int ch   = (chain_enc[node] == chain_enc[(size_t)b * LL + j]) ? 1 : 0;

    auto put_chan = [&](int n, int s) {
        int d = ch ? min(max(s + 32, 0), 64) : 65;
        const float* pw = pos_W + d * 16;
        #pragma unroll
        for (int c = 0; c < 16; ++c) xrow[n * 16 + c] = pw[c] + pos_b[c];
    };

    if (half == 0) {
        float om = omega[pair], th = theta[pair];
        put_chan(0, ri_i - ri_j);
        put_chan(1, (int)cosf(om));
        put_chan(2, (int)sinf(om));
        put_chan(3, (int)cosf(th));
    } else {
        float th = theta[pair], ph = phi[pair];
        put_chan(4, (int)sinf(th));
        put_chan(5, (int)cosf(ph));
        put_chan(6, (int)sinf(ph));
        float dn = dn_ws[e];
        #pragma unroll
        for (int c = 0; c < 16; ++c) {
            float mu = 2.0f + (20.0f / 15.0f) * (float)c;
            float rr = (dn - mu) * 0.8f;          // /1.25
            xrow[112 + c] = expf(-rr * rr);
        }
    }
    __syncthreads();

    // WMMA: A = x[16 edges x 128], B = edge_W[128 x 128], 8 N-tiles, K step 4.
    // A layout: lane m holds row m; VGPR0/1 = K0/K1 (lanes 0-15), K2/K3 (16-31).
    v8f acc[8];
    #pragma unroll
    for (int nt = 0; nt < 8; ++nt) acc[nt] = (v8f)0.0f;

    const float* __restrict__ xr = &xs[(w * 16 + m) * XPITCH];
    const int koff = 2 * half;

    for (int kc = 0; kc < 32; ++kc) {
        v2f a;
        a[0] = xr[4 * kc + koff + 0];
        a[1] = xr[4 * kc + koff + 1];
        #pragma unroll
        for (int nt = 0; nt < 8; ++nt) {
            const float* wb = &Wt[(nt * 16 + m) * XPITCH + 4 * kc + koff];
            v2f bmat;
            bmat[0] = wb[0];
            bmat[1] = wb[1];
            acc[nt] = __builtin_amdgcn_wmma_f32_16x16x4_f32(
                false, a, false, bmat, (short)0, acc[nt], false, false);
        }
    }

    // LayerNorm epilogue: stats per edge via half-wave (width=16) reductions.
    float gv[8], bvv[8];
    #pragma unroll
    for (int nt = 0; nt < 8; ++nt) {
        gv[nt]  = ln_g[nt * 16 + m];
        bvv[nt] = ln_b[nt * 16 + m];
    }

    #pragma unroll
    for (int r = 0; r < 8; ++r) {
        float s1 = 0.0f, s2 = 0.0f;
        #pragma unroll
        for (int nt = 0; nt < 8; ++nt) {
            float v = acc[nt][r];
            s1 += v;
            s2 += v * v;
        }
        #pragma unroll
        for (int o = 1; o < 16; o <<= 1) {
            s1 += __shfl_xor(s1, o, 16);
            s2 += __shfl_xor(s2, o, 16);
        }
        float mean = s1 * (1.0f / 128.0f);
        float var  = s2 * (1.0f / 128.0f) - mean * mean;
        float rs   = rsqrtf(var + 1e-5f);
        size_t edge = (size_t)e_base + r + 8 * half;   // C row M=r (lanes 0-15), M=r+8 (16-31)
        float* __restrict__ orow = outE + edge * EF_;
        #pragma unroll
        for (int nt = 0; nt < 8; ++nt)
            orow[nt * 16 + m] = (acc[nt][r] - mean) * rs * gv[nt] + bvv[nt];
    }
}

// ---------------------------------------------------------------------------
// Kernel 3: node features: [embed(S), dihedral] (12) @ node_W (12x128) + b -> LN
// One 128-thread block per node.
// ---------------------------------------------------------------------------
__global__ __launch_bounds__(128) void pf_node_kernel(
    const float* __restrict__ dihedral, const int* __restrict__ S,
    const float* __restrict__ embed_tab,
    const float* __restrict__ node_W, const float* __restrict__ node_b,
    const float* __restrict__ ln_g, const float* __restrict__ ln_b,
    float* __restrict__ outV)
{
    __shared__ float sx[12];
    __shared__ float p1[4], p2[4];
    const int node = blockIdx.x;
    const int tid  = threadIdx.x;
    const int wid  = tid >> 5;
    const int lane = tid & 31;

    if (tid < 6)       sx[tid] = embed_tab[S[node] * 6 + tid];
    else if (tid < 12) sx[tid] = dihedral[(size_t)node * 6 + (tid - 6)];
    __syncthreads();

    float y = node_b[tid];
    #pragma unroll
    for (int k = 0; k < 12; ++k) y += sx[k] * node_W[k * EF_ + tid];

    float s1 = y, s2 = y * y;
    #pragma unroll
    for (int o = 16; o >= 1; o >>= 1) {
        s1 += __shfl_xor(s1, o, 32);
        s2 += __shfl_xor(s2, o, 32);
    }
    if (lane == 0) { p1[wid] = s1; p2[wid] = s2; }
    __syncthreads();
    s1 = p1[0] + p1[1] + p1[2] + p1[3];
    s2 = p2[0] + p2[1] + p2[2] + p2[3];
    float mean = s1 * (1.0f / 128.0f);
    float var  = s2 * (1.0f / 128.0f) - mean * mean;
    float rs   = rsqrtf(var + 1e-5f);
    outV[(size_t)node * EF_ + tid] = (y - mean) * rs * ln_g[tid] + ln_b[tid];
}

// ---------------------------------------------------------------------------
extern "C" void kernel_launch(void* const* d_in, const int* in_sizes, int n_in,
                              void* d_out, int out_size, void* d_ws, size_t ws_size,
                              hipStream_t stream) {
    const float* dist        = (const float*)d_in[0];
    const float* omega       = (const float*)d_in[1];
    const float* theta       = (const float*)d_in[2];
    const float* phi         = (const float*)d_in[3];
    const float* dihedral    = (const float*)d_in[4];
    const float* mask_angle  = (const float*)d_in[5];
    // d_in[6] = mask (unused: fully masked-in setup handled via mask_angle path)
    const int*   S           = (const int*)d_in[7];
    const int*   residue_idx = (const int*)d_in[8];
    const int*   chain_enc   = (const int*)d_in[9];
    const float* pos_W       = (const float*)d_in[10];
    const float* pos_b       = (const float*)d_in[11];
    const float* edge_W      = (const float*)d_in[12];
    const float* ln_e_g      = (const float*)d_in[13];
    const float* ln_e_b      = (const float*)d_in[14];
    const float* embed_tab   = (const float*)d_in[15];
    const float* node_W      = (const float*)d_in[16];
    const float* node_b      = (const float*)d_in[17];
    const float* ln_n_g      = (const float*)d_in[18];
    const float* ln_n_b      = (const float*)d_in[19];

    const size_t nNode = (size_t)BB * LL;             // 8192
    const size_t nEdge = nNode * KNN_;                // 262144

    float* outV   = (float*)d_out;                    // [B,L,128]
    float* outE   = outV + nNode * EF_;               // [B,L,K,128]
    float* outIdx = outE + nEdge * EF_;               // [B,L,K] (as float)

    int*   idx_ws = (int*)d_ws;                       // nEdge ints
    float* dn_ws  = (float*)((char*)d_ws + nEdge * sizeof(int));

    pf_topk_kernel<<<(int)nNode, 256, 0, stream>>>(dist, mask_angle, outIdx, idx_ws, dn_ws);

    const int edgeBlocks = (int)(nEdge / 16 / 4);     // 4096 (4 waves x 16 edges)
    pf_edge_kernel<<<edgeBlocks, 128, 0, stream>>>(
        omega, theta, phi, residue_idx, chain_enc,
        pos_W, pos_b, edge_W, ln_e_g, ln_e_b,
        idx_ws, dn_ws, outE);

    pf_node_kernel<<<(int)nNode, 128, 0, stream>>>(
        dihedral, S, embed_tab, node_W, node_b, ln_n_g, ln_n_b, outV);
}